// _SparseTorch_33706903339706
// MI455X (gfx1250) — compile-verified
//
#include <hip/hip_runtime.h>

typedef __attribute__((ext_vector_type(2))) float v2f;
typedef __attribute__((ext_vector_type(8))) float v8f;

#define NV   50000
#define BSZ  64
#define DIN  1024
#define KF   32
#define DOUT 256

// ---------------------------------------------------------------------------
// Kernel 1: h[n][b] = relu( sum_d x[b][d]*W0[n][d] + b0[n] )   (h is [N][64])
// One wave per 16-row N tile; wave computes 16x64 via 4 accumulators using
// V_WMMA_F32_16X16X4_F32.  W0 streamed from HBM exactly once (204.8 MB) with
// non-temporal loads so the 192MB L2 isn't swept; x (256KB) stays temporal.
// f32 A 16x4 layout: lanes 0-15 hold (K=0,K=1) of row M=lane, lanes 16-31 hold
// (K=2,K=3) of row M=lane-16  -> one b64 load per lane at row*stride + k + 2*(lane>>4).
// B 4x16 mirrors this with lane = column.
// ---------------------------------------------------------------------------
__global__ __launch_bounds__(128) void k_layer0(
    const float* __restrict__ x, const float* __restrict__ W0,
    const float* __restrict__ b0, float* __restrict__ h) {
  const int lane = threadIdx.x & 31;
  const int wave = threadIdx.x >> 5;
  const int n0   = (blockIdx.x * 4 + wave) * 16;
  if (n0 >= NV) return;                     // uniform per wave: EXEC stays all-1s
  const int row = lane & 15;
  const int kh  = (lane >> 4) * 2;          // 0 or 2

  const float* ap  = W0 + (size_t)(n0 + row) * DIN + kh;
  const float* xp0 = x  + (size_t)(row +  0) * DIN + kh;
  const float* xp1 = x  + (size_t)(row + 16) * DIN + kh;
  const float* xp2 = x  + (size_t)(row + 32) * DIN + kh;
  const float* xp3 = x  + (size_t)(row + 48) * DIN + kh;

  v8f c0 = {}, c1 = {}, c2 = {}, c3 = {};
#pragma unroll 4
  for (int k = 0; k < DIN; k += 4) {
    v2f a   = __builtin_nontemporal_load((const v2f*)(ap + k));   // one-shot W0 stream
    v2f b0v = *(const v2f*)(xp0 + k);
    v2f b1v = *(const v2f*)(xp1 + k);
    v2f b2v = *(const v2f*)(xp2 + k);
    v2f b3v = *(const v2f*)(xp3 + k);
    c0 = __builtin_amdgcn_wmma_f32_16x16x4_f32(false, a, false, b0v, (short)0, c0, false, false);
    c1 = __builtin_amdgcn_wmma_f32_16x16x4_f32(false, a, false, b1v, (short)0, c1, false, false);
    c2 = __builtin_amdgcn_wmma_f32_16x16x4_f32(false, a, false, b2v, (short)0, c2, false, false);
    c3 = __builtin_amdgcn_wmma_f32_16x16x4_f32(false, a, false, b3v, (short)0, c3, false, false);
  }
  // C/D layout: VGPR r holds M = r (lanes 0-15) or M = 8+r (lanes 16-31); N = lane&15
  const int bl = lane & 15;
  const int mh = (lane >> 4) * 8;
#pragma unroll
  for (int r = 0; r < 8; ++r) {
    const int m = mh + r;
    const float bias = b0[n0 + m];
    float* hp = h + (size_t)(n0 + m) * BSZ;
    hp[ 0 + bl] = fmaxf(c0[r] + bias, 0.f);
    hp[16 + bl] = fmaxf(c1[r] + bias, 0.f);
    hp[32 + bl] = fmaxf(c2[r] + bias, 0.f);
    hp[48 + bl] = fmaxf(c3[r] + bias, 0.f);
  }
}

// ---------------------------------------------------------------------------
// Kernel 2 (x3): h_new[n][:] = relu( sum_k Wv[n][k] * h[src[n][k]][:] )
// One wave per vertex: lane covers 2 of the 64 batch columns, each gathered row
// is one coalesced 256B wave read from L2-resident h (temporal).  Wv/src are
// one-shot streams -> non-temporal.  Weights/indices broadcast via v_readlane.
// ---------------------------------------------------------------------------
__global__ __launch_bounds__(256) void k_gather(
    const float* __restrict__ hin, const float* __restrict__ wv,
    const long long* __restrict__ src, float* __restrict__ hout) {
  const int lane = threadIdx.x & 31;
  const int n    = blockIdx.x * 8 + (threadIdx.x >> 5);   // 6250*8 == 50000 exact
  const int wbits = (int)__float_as_uint(
      __builtin_nontemporal_load(wv + (size_t)n * KF + lane));
  const int sval  = (int)__builtin_nontemporal_load(src + (size_t)n * KF + lane);
  float ax = 0.f, ay = 0.f;
#pragma unroll 8
  for (int k = 0; k < KF; ++k) {
    const int   sk = __builtin_amdgcn_readlane(sval, k);
    const float wk = __uint_as_float((unsigned)__builtin_amdgcn_readlane(wbits, k));
    const v2f r = *(const v2f*)(hin + (size_t)sk * BSZ + lane * 2);
    ax = fmaf(wk, r[0], ax);
    ay = fmaf(wk, r[1], ay);
  }
  v2f o; o[0] = fmaxf(ax, 0.f); o[1] = fmaxf(ay, 0.f);
  *(v2f*)(hout + (size_t)n * BSZ + lane * 2) = o;
}

// ---------------------------------------------------------------------------
// Kernel 3a: out[b][d] = bout[d]   (pre-init so split-K partials can atomically add)
// ---------------------------------------------------------------------------
__global__ __launch_bounds__(256) void k_init_out(
    const float* __restrict__ bout, float* __restrict__ out) {
  const int i = blockIdx.x * 256 + threadIdx.x;
  out[i] = bout[i & (DOUT - 1)];
}

// ---------------------------------------------------------------------------
// Kernel 3b: out[b][d] += sum_n h[n][b] * Wout[d][n]
// WMMA f32 16x16x4, split-K (blockIdx.z = 50 splits of 1000), one wave per
// 16x16 output tile per split; partials accumulated with global_atomic_add_f32.
// Wout streamed once (51.2 MB, non-temporal so it doesn't evict h); h from L2.
// ---------------------------------------------------------------------------
__global__ __launch_bounds__(32) void k_out_gemm(
    const float* __restrict__ h, const float* __restrict__ Wout,
    float* __restrict__ out) {
  const int lane = threadIdx.x;
  const int dt = blockIdx.x;               // 16 tiles over D_OUT
  const int mt = blockIdx.y;               // 4 tiles over B
  const int ks = blockIdx.z * 1000;
  const int ke = ks + 1000;                // 50*1000 == 50000 exact
  const int row = lane & 15;
  const int kh  = (lane >> 4) * 2;
  const float* hp = h + (mt * 16 + row);                         // A[m][k] = h[k][b]
  const float* wp = Wout + (size_t)(dt * 16 + row) * NV + kh;    // B[k][d] = Wout[d][k]
  v8f c = {};
  for (int k = ks; k < ke; k += 4) {
    v2f a;
    a[0] = hp[(size_t)(k + kh)     * BSZ];
    a[1] = hp[(size_t)(k + kh + 1) * BSZ];
    v2f b = __builtin_nontemporal_load((const v2f*)(wp + k));    // one-shot Wout stream
    c = __builtin_amdgcn_wmma_f32_16x16x4_f32(false, a, false, b, (short)0, c, false, false);
  }
  const int bl = lane & 15;
  const int mh = (lane >> 4) * 8;
#pragma unroll
  for (int r = 0; r < 8; ++r) {
    const int b = mt * 16 + mh + r;
    const int d = dt * 16 + bl;
    atomicAdd(out + (size_t)b * DOUT + d, c[r]);
  }
}

// ---------------------------------------------------------------------------
extern "C" void kernel_launch(void* const* d_in, const int* in_sizes, int n_in,
                              void* d_out, int out_size, void* d_ws, size_t ws_size,
                              hipStream_t stream) {
  const float*     x    = (const float*)d_in[0];
  const float*     W0   = (const float*)d_in[1];
  const float*     b0   = (const float*)d_in[2];
  const float*     Wv   = (const float*)d_in[3];
  const float*     Wout = (const float*)d_in[4];
  const float*     bout = (const float*)d_in[5];
  const long long* src  = (const long long*)d_in[6];
  float* out = (float*)d_out;

  float* hA = (float*)d_ws;                    // 50000*64*4 = 12.8 MB
  float* hB = hA + (size_t)NV * BSZ;           // second 12.8 MB

  // Layer 0 GEMM+relu -> hA
  k_layer0<<<(3125 + 3) / 4, 128, 0, stream>>>(x, W0, b0, hA);

  // 3 gather layers, ping-pong hA <-> hB
  const float* hin = hA;
  float* hout = hB;
  for (int l = 0; l < 3; ++l) {
    k_gather<<<NV / 8, 256, 0, stream>>>(hin, Wv + (size_t)l * NV * KF,
                                         src + (size_t)l * NV * KF, hout);
    float* t = (float*)hin; hin = hout; hout = t;
  }

  // Output GEMM: init with bias, then split-K WMMA accumulation
  k_init_out<<<(BSZ * DOUT) / 256, 256, 0, stream>>>(bout, out);
  k_out_gemm<<<dim3(16, 4, 50), 32, 0, stream>>>(hin, Wout, out);
}